// CenterLoss_34600256537339
// MI455X (gfx1250) — compile-verified
//
#include <hip/hip_runtime.h>
#include <hip/hip_bf16.h>

// CenterLoss for MI455X (gfx1250, wave32).
// Bandwidth-bound (16.8 MB @ 23.3 TB/s ~ 0.7us). One wave per row:
// 32 lanes x 16 f32 = 512 = D. Coalesced b128 loads, f32 FMA, then a
// cross-lane reduction done on the matrix unit via V_WMMA_F32_16X16X4_F32
// (A = per-lane partials in the 16x4 A layout, B = ones). Deterministic
// two-kernel reduction (no float atomics), scratch fully rewritten per call.

typedef __attribute__((ext_vector_type(2))) float v2f;
typedef __attribute__((ext_vector_type(8))) float v8f;

#define D_DIM 512
#define LAMBDA_C 0.003f
#define CLAMP_MIN 1e-12f
#define CLAMP_MAX 1e12f

// Sum a single f32 value across all 32 lanes using the WMMA f32 16x16x4 unit.
// A layout (32-bit A, 16x4): lanes 0-15 VGPR0 = A[m,0], lanes 16-31 VGPR0 = A[m,2].
// With A.y = 0 and B = ones, D[m,n] = p[m] + p[m+16] = s[m].
// D layout: VGPR j, lanes 0-15 hold s[j], lanes 16-31 hold s[j+8].
// Sum of the 8 D VGPRs gives sum(s[0..7]) in lanes 0-15 and sum(s[8..15]) in
// lanes 16-31; one xor-16 shuffle finishes the reduction (result in all lanes).
__device__ __forceinline__ float wave_sum_f32_wmma(float p) {
    v2f a; a.x = p;   a.y = 0.0f;
    v2f b; b.x = 1.0f; b.y = 1.0f;
    v8f c = {};
    v8f d = __builtin_amdgcn_wmma_f32_16x16x4_f32(
        /*neg_a=*/false, a, /*neg_b=*/false, b,
        /*c_mod=*/(short)0, c, /*reuse_a=*/false, /*reuse_b=*/false);
    float t = ((d[0] + d[1]) + (d[2] + d[3])) + ((d[4] + d[5]) + (d[6] + d[7]));
    t += __shfl_xor(t, 16, 32);
    return t;
}

__global__ void __launch_bounds__(256)
center_loss_rows(const float* __restrict__ emb,
                 const int*   __restrict__ labels,
                 const float* __restrict__ centers,
                 float* __restrict__ row_dist,
                 int B) {
    const int wave = (int)((blockIdx.x * blockDim.x + threadIdx.x) >> 5);
    const int lane = (int)(threadIdx.x & 31);
    if (wave >= B) return;  // wave-uniform: EXEC stays all-1s for live waves

    const float* __restrict__ e = emb + (size_t)wave * D_DIM;
    const float* __restrict__ c = centers + (size_t)labels[wave] * D_DIM;

    float acc = 0.0f;
#pragma unroll
    for (int j = 0; j < 4; ++j) {
        const int off = lane * 4 + j * 128;          // 512B contiguous per wave per load
        const float4 ev = *(const float4*)(e + off); // global_load_b128
        const float4 cv = *(const float4*)(c + off);
        float dx;
        dx = ev.x - cv.x; acc = fmaf(dx, dx, acc);
        dx = ev.y - cv.y; acc = fmaf(dx, dx, acc);
        dx = ev.z - cv.z; acc = fmaf(dx, dx, acc);
        dx = ev.w - cv.w; acc = fmaf(dx, dx, acc);
    }

    const float dist = wave_sum_f32_wmma(acc);       // full-row ||e-c||^2
    if (lane == 0) {
        row_dist[wave] = fminf(fmaxf(dist, CLAMP_MIN), CLAMP_MAX);
    }
}

__global__ void __launch_bounds__(256)
center_loss_finalize(const float* __restrict__ row_dist,
                     float* __restrict__ out,
                     int B, int C) {
    __shared__ float s[256];
    float t = 0.0f;
    for (int i = (int)threadIdx.x; i < B; i += 256) t += row_dist[i]; // fixed order
    s[threadIdx.x] = t;
    __syncthreads();
#pragma unroll
    for (int k = 128; k > 0; k >>= 1) {
        if ((int)threadIdx.x < k) s[threadIdx.x] += s[threadIdx.x + k];
        __syncthreads();
    }
    if (threadIdx.x == 0) {
        const float masked_floor = (float)((double)B * (double)(C - 1) * 1e-12);
        out[0] = LAMBDA_C * ((s[0] + masked_floor) / (float)B);
    }
}

extern "C" void kernel_launch(void* const* d_in, const int* in_sizes, int n_in,
                              void* d_out, int out_size, void* d_ws, size_t ws_size,
                              hipStream_t stream) {
    (void)n_in; (void)out_size; (void)ws_size;
    const float* emb     = (const float*)d_in[0];   // [B, 512] f32
    const int*   labels  = (const int*)d_in[1];     // [B] int
    const float* centers = (const float*)d_in[2];   // [C, 512] f32
    float*       out     = (float*)d_out;           // scalar f32
    float*       rowbuf  = (float*)d_ws;            // B floats (16 KB), fully rewritten

    const int B = in_sizes[0] / D_DIM;              // 4096
    const int C = in_sizes[2] / D_DIM;              // 10000

    const int waves_per_block = 8;                  // 256 threads = 8 wave32
    const int threads = waves_per_block * 32;
    const int blocks  = (B + waves_per_block - 1) / waves_per_block;

    center_loss_rows<<<blocks, threads, 0, stream>>>(emb, labels, centers, rowbuf, B);
    center_loss_finalize<<<1, 256, 0, stream>>>(rowbuf, out, B, C);
}